// Attention_28991029248776
// MI455X (gfx1250) — compile-verified
//
#include <hip/hip_runtime.h>
#include <math.h>

typedef __attribute__((ext_vector_type(2))) float v2f;
typedef __attribute__((ext_vector_type(8))) float v8f;

#define B_ 4
#define T_ 512
#define H_ 256
#define SCHUNK 32  // score columns per block in the scores kernel

__device__ __forceinline__ v8f wmma_f32_16x16x4(v2f a, v2f b, v8f c) {
  // 8 args: (neg_a, A, neg_b, B, c_mod, C, reuse_a, reuse_b)
  return __builtin_amdgcn_wmma_f32_16x16x4_f32(false, a, false, b, (short)0, c,
                                               false, false);
}

// Hardware transcendentals on gfx1250 when available.
#if __has_builtin(__builtin_amdgcn_tanhf)
__device__ __forceinline__ float fast_tanh(float x) {
  return __builtin_amdgcn_tanhf(x);  // v_tanh_f32
}
#else
__device__ __forceinline__ float fast_tanh(float x) { return tanhf(x); }
#endif

#if __has_builtin(__builtin_amdgcn_exp2f)
__device__ __forceinline__ float fast_exp(float x) {
  return __builtin_amdgcn_exp2f(x * 1.4426950408889634f);  // v_exp_f32
}
#else
__device__ __forceinline__ float fast_exp(float x) { return expf(x); }
#endif

// ---------------------------------------------------------------------------
// Kernel 1: Q = X @ W1^T, K = X @ W2^T via V_WMMA_F32_16X16X4_F32.
// One wave per 16x16 output tile; Q and K share the A fragment.
// ---------------------------------------------------------------------------
__global__ __launch_bounds__(256) void proj_qk_kernel(
    const float* __restrict__ X, const float* __restrict__ W1,
    const float* __restrict__ W2, float* __restrict__ Q, float* __restrict__ K) {
  const int lane = threadIdx.x & 31;
  const int wave = threadIdx.x >> 5;
  const int gw   = blockIdx.x * 8 + wave;  // 2048 waves = 128 row * 16 h tiles
  const int row0 = (gw >> 4) * 16;
  const int h0   = (gw & 15) * 16;
  const int half = lane >> 4;  // 0: K=0,1   1: K=2,3
  const int lrow = lane & 15;

  const float* __restrict__ xrow  = X  + (size_t)(row0 + lrow) * H_;
  const float* __restrict__ w1row = W1 + (size_t)(h0 + lrow) * H_;
  const float* __restrict__ w2row = W2 + (size_t)(h0 + lrow) * H_;

  v8f cq = {};
  v8f ck = {};
  for (int j = 0; j < H_; j += 4) {
    const int kb = j + 2 * half;
    v2f a, b1, b2;
    a.x  = xrow[kb];   a.y  = xrow[kb + 1];
    b1.x = w1row[kb];  b1.y = w1row[kb + 1];
    b2.x = w2row[kb];  b2.y = w2row[kb + 1];
    cq = wmma_f32_16x16x4(a, b1, cq);
    ck = wmma_f32_16x16x4(a, b2, ck);
  }

#pragma unroll
  for (int i = 0; i < 8; ++i) {
    const int m = i + 8 * half;  // C/D layout: lanes 16-31 hold M=8..15
    Q[(size_t)(row0 + m) * H_ + h0 + lrow] = cq[i];
    K[(size_t)(row0 + m) * H_ + h0 + lrow] = ck[i];
  }
}

// ---------------------------------------------------------------------------
// Kernel 2: raw scores S[b,t,s] = sum_h v[h]*tanh(q[t,h]+k[s,h]) for s<=t.
// Grid: (128 t-tiles) x (16 s-chunks of 32); non-causal chunks exit early.
// Raw scores are written into the weights region of d_out (scratch), then
// normalized in place by kernel 3.
// ---------------------------------------------------------------------------
__global__ __launch_bounds__(256) void scores_kernel(
    const float* __restrict__ Q, const float* __restrict__ K,
    const float* __restrict__ v, float* __restrict__ Sraw) {
  const int b    = blockIdx.x >> 5;
  const int t0   = (blockIdx.x & 31) * 16;
  const int s0   = blockIdx.y * SCHUNK;
  const int nk   = t0 + 16;  // causal extent of this tile
  if (s0 >= nk) return;

  __shared__ float qs[16][H_];
  const int lane = threadIdx.x & 31;
  const int wave = threadIdx.x >> 5;
  const size_t rowBase = (size_t)b * T_;

  float vf[8];
#pragma unroll
  for (int i = 0; i < 8; ++i) vf[i] = v[i * 32 + lane];

  for (int idx = threadIdx.x; idx < 16 * H_; idx += 256) {
    const int t = idx >> 8, h = idx & (H_ - 1);
    qs[t][h] = Q[(rowBase + t0 + t) * H_ + h];
  }
  __syncthreads();

  float* __restrict__ srow = Sraw + ((size_t)b * T_ + t0) * T_;
  for (int sc = wave; sc < SCHUNK; sc += 8) {
    const int s = s0 + sc;
    if (s >= nk) continue;  // column entirely masked (never read later)
    float kf[8];
    const float* __restrict__ krow = K + (rowBase + s) * H_;
#pragma unroll
    for (int i = 0; i < 8; ++i) kf[i] = krow[i * 32 + lane];

    for (int t = 0; t < 16; ++t) {
      float acc;
      if (s <= t0 + t) {
        acc = 0.f;
#pragma unroll
        for (int i = 0; i < 8; ++i)
          acc += vf[i] * fast_tanh(qs[t][i * 32 + lane] + kf[i]);
#pragma unroll
        for (int off = 16; off >= 1; off >>= 1)
          acc += __shfl_xor(acc, off, 32);
      } else {
        acc = -INFINITY;  // masked entries inside the diagonal chunk
      }
      if (lane == 0) srow[(size_t)t * T_ + s] = acc;
    }
  }
}

// ---------------------------------------------------------------------------
// Kernel 3: per 16-row tile: softmax rows in LDS, write normalized weights
// (zeros for future positions), then context = P(16 x nk) @ X(nk x 256)
// via f32 WMMA.
// ---------------------------------------------------------------------------
__global__ __launch_bounds__(256) void softmax_ctx_kernel(
    const float* __restrict__ X, float* __restrict__ ctx,
    float* __restrict__ wts) {
  __shared__ float S[16][T_ + 4];  // +4 pad: conflict-free WMMA A-frag reads

  const int lane = threadIdx.x & 31;
  const int wave = threadIdx.x >> 5;
  const int b    = blockIdx.x >> 5;
  const int t0   = (blockIdx.x & 31) * 16;
  const size_t rowBase = (size_t)b * T_;
  const int nk   = t0 + 16;

  float* __restrict__ wrow = wts + ((size_t)b * T_ + t0) * T_;

  // Load raw scores (written by scores_kernel) for the causal region.
  for (int t = 0; t < 16; ++t)
    for (int s = threadIdx.x; s < nk; s += 256)
      S[t][s] = wrow[(size_t)t * T_ + s];
  __syncthreads();

  // Row softmax: each wave owns rows wave and wave+8.
  for (int rr = 0; rr < 2; ++rr) {
    const int t = wave + rr * 8;
    float m = -INFINITY;
    for (int s = lane; s < nk; s += 32) m = fmaxf(m, S[t][s]);
#pragma unroll
    for (int off = 16; off >= 1; off >>= 1)
      m = fmaxf(m, __shfl_xor(m, off, 32));
    float sum = 0.f;
    for (int s = lane; s < nk; s += 32) {
      const float e = fast_exp(S[t][s] - m);  // exp(-inf)=0 handles the mask
      S[t][s] = e;
      sum += e;
    }
#pragma unroll
    for (int off = 16; off >= 1; off >>= 1)
      sum += __shfl_xor(sum, off, 32);
    const float inv = 1.0f / sum;  // row always has >=1 valid entry
    for (int s = lane; s < nk; s += 32) S[t][s] *= inv;
  }
  __syncthreads();

  // Write normalized weights; future columns are exactly 0.
  for (int idx = threadIdx.x; idx < 16 * T_; idx += 256) {
    const int t = idx >> 9, s = idx & (T_ - 1);
    wrow[(size_t)t * T_ + s] = (s < nk) ? S[t][s] : 0.f;
  }

  // Context GEMM: P(16 x nk) @ X(nk x 256) via f32 WMMA.
  const int half = lane >> 4;
  const int lrow = lane & 15;
  for (int hb = wave; hb < 16; hb += 8) {
    const int h0 = hb * 16;
    v8f c = {};
    for (int k = 0; k < nk; k += 4) {
      const int kb = k + 2 * half;
      v2f a, bb;
      a.x = S[lrow][kb];  // A frag from probs in LDS
      a.y = S[lrow][kb + 1];
      const float* __restrict__ x0 = X + (rowBase + kb) * H_ + h0 + lrow;
      bb.x = x0[0];       // B frag: X rows kb, kb+1
      bb.y = x0[H_];
      c = wmma_f32_16x16x4(a, bb, c);
    }
#pragma unroll
    for (int i = 0; i < 8; ++i)
      ctx[(rowBase + t0 + i + 8 * half) * H_ + h0 + lrow] = c[i];
  }
}

// ---------------------------------------------------------------------------
extern "C" void kernel_launch(void* const* d_in, const int* in_sizes, int n_in,
                              void* d_out, int out_size, void* d_ws,
                              size_t ws_size, hipStream_t stream) {
  (void)in_sizes; (void)n_in; (void)out_size; (void)ws_size;
  const float* X  = (const float*)d_in[0];  // (B,T,H)
  const float* W1 = (const float*)d_in[1];  // (H,H)
  const float* W2 = (const float*)d_in[2];  // (H,H)
  const float* v  = (const float*)d_in[3];  // (H,)

  float* Q   = (float*)d_ws;                  // 2 MB
  float* K   = Q + (size_t)B_ * T_ * H_;      // 2 MB
  float* ctx = (float*)d_out;                 // (B,T,H)
  float* wts = ctx + (size_t)B_ * T_ * H_;    // (B,T,T); also scores scratch

  proj_qk_kernel<<<256, 256, 0, stream>>>(X, W1, W2, Q, K);
  dim3 sgrid(B_ * (T_ / 16), T_ / SCHUNK);
  scores_kernel<<<sgrid, 256, 0, stream>>>(Q, K, v, wts);
  softmax_ctx_kernel<<<B_ * (T_ / 16), 256, 0, stream>>>(X, ctx, wts);
}